// GCN2_1709396984303
// MI455X (gfx1250) — compile-verified
//
#include <hip/hip_runtime.h>
#include <math.h>

// ---------------- problem constants (match reference) ----------------
#define NNODES 50000
#define NEDGES 800000
#define HID    128
#define OUT_DIM 64
#define NUM_CONVS 6

#define ALPHA_C  0.1f
#define ONE_MA   0.9f
// beta = log(0.5/9 + 1) = ln(19/18)
#define BETA_C   0.0540672212702758f
#define ONE_MB   0.9459327787297242f

typedef __attribute__((ext_vector_type(2))) float v2f;
typedef __attribute__((ext_vector_type(8))) float v8f;

// ---------------------------------------------------------------------
// Fused GEMM on native f32 WMMA (V_WMMA_F32_16X16X4_F32).
// Block: (NCOLS/16) waves * 32 threads; block computes 16 rows x NCOLS.
// W is staged TRANSPOSED in LDS with padded stride so both A and B
// fragments are single ds_load_b64 ops with conflict-free banking.
// MODE 0: out = relu(A@W + bias)
// MODE 1: z = (1-a)*A + a*Aux;  out = relu((1-b)*z + b*(z@W))   (in-place ok)
// MODE 2: out = A@W + bias
// ---------------------------------------------------------------------
template<int NCOLS, int MODE>
__global__ void gemm_k(const float* __restrict__ A,
                       const float* __restrict__ Aux,
                       const float* __restrict__ W,
                       const float* __restrict__ bias,
                       float* __restrict__ out)
{
    constexpr int NW   = NCOLS / 16;     // waves per block
    constexpr int NT   = NW * 32;        // threads per block
    constexpr int ZS   = HID + 4;        // padded A-panel stride
    constexpr int WT_S = HID + 4;        // padded W^T row stride (per output col)

    __shared__ float sWt[NCOLS * WT_S];  // sWt[n*WT_S + k] = W[k][n]
    __shared__ float sZ[16 * ZS];        // 16-row A/z panel

    const int tid  = threadIdx.x;
    const int row0 = blockIdx.x * 16;

    // ---- stage W transposed into LDS (coalesced global reads) ----
    for (int i = tid; i < HID * NCOLS; i += NT) {
        const int k = i / NCOLS;
        const int n = i % NCOLS;
        sWt[n * WT_S + k] = W[i];
    }

    // ---- build the 16 x HID A-panel (and z for MODE 1) ----
    for (int i = tid * 4; i < 16 * HID; i += NT * 4) {
        const int r = i / HID;
        const int c = i % HID;
        float4 a = *(const float4*)&A[(size_t)(row0 + r) * HID + c];
        if (MODE == 1) {
            const float4 x0 = *(const float4*)&Aux[(size_t)(row0 + r) * HID + c];
            a.x = a.x * ONE_MA + ALPHA_C * x0.x;
            a.y = a.y * ONE_MA + ALPHA_C * x0.y;
            a.z = a.z * ONE_MA + ALPHA_C * x0.z;
            a.w = a.w * ONE_MA + ALPHA_C * x0.w;
        }
        *(float4*)&sZ[r * ZS + c] = a;   // r*ZS floats = 528 B/row, 16B aligned
    }
    __syncthreads();

    const int wave   = tid >> 5;
    const int lane   = tid & 31;
    const int col0   = wave * 16;
    const int laneLo = lane & 15;        // M (A) / N (B)
    const int laneHi = lane >> 4;        // selects K pair {0,1} vs {2,3}

    const float* zrow = &sZ [laneLo * ZS];
    const float* wcol = &sWt[(col0 + laneLo) * WT_S];

    v8f acc = {};
    #pragma unroll
    for (int k = 0; k < HID; k += 4) {
        const int kb = k + laneHi * 2;
        v2f a; a.x = zrow[kb]; a.y = zrow[kb + 1];   // one ds_load_b64
        v2f b; b.x = wcol[kb]; b.y = wcol[kb + 1];   // one ds_load_b64
        acc = __builtin_amdgcn_wmma_f32_16x16x4_f32(
            false, a, false, b, (short)0, acc, false, false);
    }

    // ---- epilogue; C/D layout: VGPR r -> M = r + 8*laneHi, N = laneLo ----
    #pragma unroll
    for (int r = 0; r < 8; ++r) {
        const int m = r + laneHi * 8;
        const int n = col0 + laneLo;
        const float v = acc[r];
        float res;
        if (MODE == 0)       res = fmaxf(v + bias[n], 0.0f);
        else if (MODE == 2)  res = v + bias[n];
        else {
            const float z = sZ[m * ZS + n];
            res = fmaxf(ONE_MB * z + BETA_C * v, 0.0f);
        }
        out[(size_t)(row0 + m) * NCOLS + n] = res;
    }
}

// ---------------------------------------------------------------------
// CSR-by-destination build (once per launch)
// ---------------------------------------------------------------------
__global__ void count_deg(const int* __restrict__ dst, int* __restrict__ counts)
{
    const int e = blockIdx.x * blockDim.x + threadIdx.x;
    if (e >= NEDGES) return;
    __hip_atomic_fetch_add(&counts[dst[e]], 1, __ATOMIC_RELAXED, __HIP_MEMORY_SCOPE_AGENT);
}

// single block of 1024 threads: exclusive scan of counts[n] -> rowptr[n+1]
__global__ void exscan_k(const int* __restrict__ counts, int* __restrict__ rowptr, int n)
{
    __shared__ int tmp[1024];
    int carry = 0;
    for (int base = 0; base < n; base += 1024) {
        const int i = base + (int)threadIdx.x;
        const int v = (i < n) ? counts[i] : 0;
        tmp[threadIdx.x] = v;
        __syncthreads();
        #pragma unroll
        for (int off = 1; off < 1024; off <<= 1) {
            const int t = (threadIdx.x >= (unsigned)off) ? tmp[threadIdx.x - off] : 0;
            __syncthreads();
            tmp[threadIdx.x] += t;
            __syncthreads();
        }
        if (i < n) rowptr[i] = carry + tmp[threadIdx.x] - v;  // exclusive
        const int total = tmp[1023];
        __syncthreads();
        carry += total;
    }
    if (threadIdx.x == 0) rowptr[n] = carry;
}

__global__ void csr_place(const int* __restrict__ src, const int* __restrict__ dst,
                          const float* __restrict__ ew,
                          int* __restrict__ rowcur,
                          int* __restrict__ csr_src, float* __restrict__ csr_w)
{
    const int e = blockIdx.x * blockDim.x + threadIdx.x;
    if (e >= NEDGES) return;
    const int d   = dst[e];
    const int pos = __hip_atomic_fetch_add(&rowcur[d], 1, __ATOMIC_RELAXED,
                                           __HIP_MEMORY_SCOPE_AGENT);
    csr_src[pos] = src[e];
    csr_w[pos]   = ew[e];
}

// ---------------------------------------------------------------------
// Gather SpMM: agg[row] = sum_j w_j * h[src_j].  One wave per dst row,
// each lane owns a float4 feature slice; register accumulation, no atomics.
// ---------------------------------------------------------------------
__global__ void spmm_csr(const float* __restrict__ h,
                         const int*   __restrict__ rowptr,
                         const int*   __restrict__ csr_src,
                         const float* __restrict__ csr_w,
                         float*       __restrict__ agg)
{
    const int gid  = blockIdx.x * blockDim.x + threadIdx.x;
    const int row  = gid >> 5;
    if (row >= NNODES) return;
    const int lane = gid & 31;

    const int beg = rowptr[row];
    const int end = rowptr[row + 1];

    float4 acc = {0.f, 0.f, 0.f, 0.f};
    for (int j = beg; j < end; ++j) {
        const int   s = csr_src[j];
        const float w = csr_w[j];
        const float4 v = *(const float4*)&h[(size_t)s * HID + lane * 4];
        acc.x += v.x * w;
        acc.y += v.y * w;
        acc.z += v.z * w;
        acc.w += v.w * w;
    }
    *(float4*)&agg[(size_t)row * HID + lane * 4] = acc;
}

// ---------------------------------------------------------------------
extern "C" void kernel_launch(void* const* d_in, const int* in_sizes, int n_in,
                              void* d_out, int out_size, void* d_ws, size_t ws_size,
                              hipStream_t stream)
{
    const float* x     = (const float*)d_in[0];   // [N, 128]
    const int*   esrc  = (const int*  )d_in[1];   // [E]
    const int*   edst  = (const int*  )d_in[2];   // [E]
    const float* ew    = (const float*)d_in[3];   // [E]
    const float* W0    = (const float*)d_in[4];   // [128, 128]
    const float* b0    = (const float*)d_in[5];   // [128]
    const float* W1    = (const float*)d_in[6];   // [128, 64]
    const float* b1    = (const float*)d_in[7];   // [64]
    const float* convw = (const float*)d_in[8];   // [6, 128, 128]
    float*       out   = (float*)d_out;           // [N, 64]
    (void)ws_size; (void)in_sizes; (void)n_in; (void)out_size;

    // ---- workspace layout ----
    const size_t bufElems = (size_t)NNODES * HID;
    char* base = (char*)d_ws;
    float* buf0    = (float*)base;                      base += bufElems * 4; // x0
    float* buf1    = (float*)base;                      base += bufElems * 4; // h/agg ping
    float* buf2    = (float*)base;                      base += bufElems * 4; // h/agg pong
    int*   rowptr  = (int*)base;                        base += (NNODES + 1) * 4;
    int*   rowcur  = (int*)base;                        base += NNODES * 4;   // also counts
    int*   csr_src = (int*)base;                        base += (size_t)NEDGES * 4;
    float* csr_w   = (float*)base;                      base += (size_t)NEDGES * 4;

    const int gemmGrid  = NNODES / 16;                  // 3125, exact
    const int edgeGrid  = (NEDGES + 255) / 256;
    const int spmmGrid  = (NNODES * 32 + 255) / 256;

    // ---- build CSR by destination (once; reused by all 6 convs) ----
    hipMemsetAsync(rowcur, 0, NNODES * sizeof(int), stream);         // counts
    count_deg<<<edgeGrid, 256, 0, stream>>>(edst, rowcur);
    exscan_k<<<1, 1024, 0, stream>>>(rowcur, rowptr, NNODES);
    hipMemcpyAsync(rowcur, rowptr, NNODES * sizeof(int),
                   hipMemcpyDeviceToDevice, stream);
    csr_place<<<edgeGrid, 256, 0, stream>>>(esrc, edst, ew, rowcur, csr_src, csr_w);

    // ---- layer 0: h = relu(x @ W0 + b0) -> buf0 (serves as x0) ----
    gemm_k<HID, 0><<<gemmGrid, 256, 0, stream>>>(x, nullptr, W0, b0, buf0);

    const float* hin = buf0;
    for (int i = 0; i < NUM_CONVS; ++i) {
        float* agg = (i % 2 == 0) ? buf1 : buf2;        // always != hin
        spmm_csr<<<spmmGrid, 256, 0, stream>>>(hin, rowptr, csr_src, csr_w, agg);
        // z = (1-a)*agg + a*x0 ; h = relu((1-b)*z + b*(z@W)); in place over agg
        gemm_k<HID, 1><<<gemmGrid, 256, 0, stream>>>(agg, buf0,
                                                     convw + (size_t)i * HID * HID,
                                                     nullptr, agg);
        hin = agg;
    }

    // ---- final: out = h @ W1 + b1 ----
    gemm_k<OUT_DIM, 2><<<gemmGrid, 128, 0, stream>>>(hin, nullptr, W1, b1, out);
}